// TemporalFilter_51685636440699
// MI455X (gfx1250) — compile-verified
//
#include <hip/hip_runtime.h>
#include <math.h>

// MI455X (gfx1250) temporal-filter kernel.
//
// Roofline: 336 MB of HBM traffic @ 23.3 TB/s ~= 14.4 us; 3.4 GFLOP of math.
// The 16 conv responses per position form a (16x12) x (12xN) GEMM; we run it
// on the matrix pipe with V_WMMA_F32_16X16X4_F32 (3 K-slices of 4 = K=12),
// leaving VALU for the g_o^2+g_e^2 epilogue, so the kernel stays HBM-bound.
// Tile staging uses the Tensor Data Mover (tensor_load_to_lds, TENSORcnt)
// for interior tiles; row-edge tiles fall back to a guarded VALU path for
// the zero-padding semantics of the 'same' convolution.

typedef float v2f __attribute__((ext_vector_type(2)));
typedef float v8f __attribute__((ext_vector_type(8)));
typedef unsigned int u32x4 __attribute__((ext_vector_type(4)));
typedef int i32x4 __attribute__((ext_vector_type(4)));
typedef int i32x8 __attribute__((ext_vector_type(8)));

#define KS      6
#define NF      8
#define NPOS    8192
#define NROWS   1024          // B*C = 16*64
#define TILE    1024          // positions per block
#define LROW    (TILE + 8)    // +2 front halo, +3 back halo, pad to 8
#define RN      ((size_t)NROWS * (size_t)NPOS)

static __device__ inline v8f wmma_k4(v2f a, v2f b, v8f c) {
    // D = A(16x4) * B(4x16) + C ;  f32 WMMA, wave32
    return __builtin_amdgcn_wmma_f32_16x16x4_f32(
        /*neg_a=*/false, a, /*neg_b=*/false, b,
        /*c_mod=*/(short)0, c, /*reuse_a=*/false, /*reuse_b=*/false);
}

static __device__ inline float swap16(float v) {
    // group-of-32 swizzle, xor_mask=0x10: exchange lane-halves of the wave
    return __int_as_float(__builtin_amdgcn_ds_swizzle(__float_as_int(v), 0x401F));
}

// TDM: 1-D tile copy of LROW f32 elements, global -> LDS.
// D# bitfields per CDNA5 ISA 8.3/8.4: group0 = {count, lds_addr, global_addr,
// type=2}; group1 = {data_size=4B, tensor_dim0, tensor_dim1=1, tile_dim0,
// tile_dim1=1, tensor_dim0_stride}; higher groups unused (dims 2..4 = 0).
static __device__ inline void tdm_load_row(unsigned lds_addr,
                                           unsigned long long ga) {
    u32x4 g0;
    g0[0] = 1u;                                         // count=1, user D#
    g0[1] = lds_addr;                                   // LDS byte offset
    g0[2] = (unsigned)(ga & 0xFFFFFFFFull);             // global_addr[31:0]
    g0[3] = (unsigned)((ga >> 32) & 0x01FFFFFFull)      // global_addr[56:32]
          | (2u << 30);                                 // type = 2 ("image")
    i32x8 g1;
    g1[0] = (int)(2u << 16);                            // data_size=2 (4B)
    g1[1] = (int)((unsigned)(LROW & 0xFFFF) << 16);     // tensor_dim0[15:0]
    g1[2] = (int)(((unsigned)LROW >> 16) | (1u << 16)); // dim0 hi | tensor_dim1=1
    g1[3] = (int)((unsigned)LROW << 16);                // tile_dim0 = LROW
    g1[4] = 1;                                          // tile_dim1=1, tile_dim2=0
    g1[5] = LROW;                                       // tensor_dim0_stride
    g1[6] = 0;
    g1[7] = 0;
    const i32x4 z4 = {0, 0, 0, 0};
    const i32x8 z8 = {0, 0, 0, 0, 0, 0, 0, 0};
    // 6-arg form on this toolchain (clang-23 / therock-10.0 headers).
    __builtin_amdgcn_tensor_load_to_lds(g0, g1, z4, z4, z8, 0);
}

__global__ __launch_bounds__(256)
void temporal_filter_wmma(const float* __restrict__ xs,
                          const float* __restrict__ xc,
                          const float* __restrict__ ft,
                          const float* __restrict__ tao,
                          float* __restrict__ out)
{
    __shared__ float s_xs[LROW];
    __shared__ float s_xc[LROW];
    __shared__ float s_w[16][12];   // rows 0..7: g_o taps, rows 8..15: g_e taps

    const int tid  = threadIdx.x;
    const int blk  = blockIdx.x;
    const int row  = blk >> 3;           // 8 tiles per row
    const int tile = blk & 7;
    const int t0   = tile * TILE;

    const float* grs = xs + (size_t)row * NPOS;
    const float* grc = xc + (size_t)row * NPOS;

    // Interior tiles: window [t0-2, t0+LROW-2) is fully in-range -> TDM DMA.
    const bool interior = (tile != 0) && (tile != 7);
    if (interior && tid < 32) {          // wave 0 issues both DMAs
        tdm_load_row((unsigned)(size_t)(void*)s_xs,
                     (unsigned long long)(size_t)(grs + t0 - 2));
        tdm_load_row((unsigned)(size_t)(void*)s_xc,
                     (unsigned long long)(size_t)(grc + t0 - 2));
    }

    // ---- 16x12 weight matrix from ft/tao (48 threads, 4 entries each) ----
    // (overlaps with the in-flight TDM transfers)
    if (tid < NF * KS) {
        const int f = tid / KS;
        const int k = tid - f * KS;
        const float fts  = 0.25f      / (1.0f + expf(-ft[f]));
        const float taos = 3.7280043f / (1.0f + expf(-tao[f]));   // -K/ln(0.2)
        const float env  = expf(-(float)k / taos);
        const float ph   = 6.2831853f * fts * (float)k;
        const float tsin = env * sinf(ph);
        const float tcos = env * cosf(ph);
        s_w[f][k]         =  tcos;   // g_o: x_sin taps
        s_w[f][k + 6]     =  tsin;   // g_o: x_cos taps
        s_w[f + 8][k]     =  tsin;   // g_e: x_sin taps
        s_w[f + 8][k + 6] = -tcos;   // g_e: x_cos taps
    }

    // Row-edge tiles: guarded scalar staging with zero padding.
    if (!interior) {
        for (int i = tid; i < TILE + 6; i += 256) {
            const int g  = t0 - 2 + i;
            const bool ok = (g >= 0) && (g < NPOS);
            s_xs[i] = ok ? grs[g] : 0.0f;
            s_xc[i] = ok ? grc[g] : 0.0f;
        }
    }

    if (interior && tid < 32) {
        __builtin_amdgcn_s_wait_tensorcnt(0);   // issuing wave drains TDM
    }
    __syncthreads();                            // publish LDS to all waves

    const int  lane = tid & 31;
    const int  wave = tid >> 5;
    const int  p    = lane & 15;
    const bool hi   = lane >= 16;

    // ---- A matrix: 16x4 f32 per K-slice.  VGPR0 = K0|K2, VGPR1 = K1|K3 ----
    v2f a0, a1, a2;
    {
        const int kb = hi ? 2 : 0;     // lanes 16-31 carry K+2 columns
        a0.x = s_w[p][0 + kb];  a0.y = s_w[p][1 + kb];
        a1.x = s_w[p][4 + kb];  a1.y = s_w[p][5 + kb];
        a2.x = s_w[p][8 + kb];  a2.y = s_w[p][9 + kb];
    }

    // im2col gather offsets for B (4x16, lane-halves = K and K+2 rows).
    // X[k][n] = k<6 ? xs[n-2+k] : xc[n-2+(k-6)];  LDS index(n) = n-t0+2.
    const int q0 = p + (hi ? 2 : 0);                // slice 0: all xs
    const int q1 = p + (hi ? 0 : 4);                // slice 1: xs (lo) / xc (hi)
    const int q2 = p + (hi ? 4 : 2);                // slice 2: all xc
    const float* b1base = hi ? s_xc : s_xs;

    const size_t obase = (size_t)row * NPOS + (size_t)t0;

    #pragma unroll
    for (int it = 0; it < 4; ++it) {
        const int t = wave * 32 + it * 256;         // 32 positions per iter

        v8f c0 = {};                                 // n in [t, t+16)
        {
            v2f b0, b1, b2;
            b0.x = s_xs[t + q0];      b0.y = s_xs[t + q0 + 1];
            b1.x = b1base[t + q1];    b1.y = b1base[t + q1 + 1];
            b2.x = s_xc[t + q2];      b2.y = s_xc[t + q2 + 1];
            c0 = wmma_k4(a0, b0, c0);
            c0 = wmma_k4(a1, b1, c0);
            c0 = wmma_k4(a2, b2, c0);
        }
        v8f c1 = {};                                 // n in [t+16, t+32)
        {
            const int u = t + 16;
            v2f b0, b1, b2;
            b0.x = s_xs[u + q0];      b0.y = s_xs[u + q0 + 1];
            b1.x = b1base[u + q1];    b1.y = b1base[u + q1 + 1];
            b2.x = s_xc[u + q2];      b2.y = s_xc[u + q2 + 1];
            c1 = wmma_k4(a0, b0, c1);
            c1 = wmma_k4(a1, b1, c1);
            c1 = wmma_k4(a2, b2, c1);
        }

        // C VGPR f holds g_o[f] (lanes 0-15) and g_e[f] (lanes 16-31):
        // me = g_o^2 + g_e^2 via lane-half swap; both halves end up valid,
        // so lane-halves carry the two 16-chunks of one coalesced store.
        const size_t nb = obase + (size_t)(t + lane);
        #pragma unroll
        for (int f = 0; f < NF; ++f) {
            const float v0 = c0[f], v1 = swap16(v0);
            const float w0 = c1[f], w1 = swap16(w0);
            const float me0 = fmaf(v0, v0, v1 * v1);
            const float me1 = fmaf(w0, w0, w1 * w1);
            out[(size_t)f * RN + nb] = hi ? me1 : me0;
        }
    }
}

extern "C" void kernel_launch(void* const* d_in, const int* in_sizes, int n_in,
                              void* d_out, int out_size, void* d_ws, size_t ws_size,
                              hipStream_t stream) {
    (void)in_sizes; (void)n_in; (void)d_ws; (void)ws_size; (void)out_size;
    const float* xs  = (const float*)d_in[0];
    const float* xc  = (const float*)d_in[1];
    const float* ft  = (const float*)d_in[2];
    const float* tao = (const float*)d_in[3];
    float* out = (float*)d_out;

    const int blocks = NROWS * (NPOS / TILE);       // 1024 rows * 8 tiles = 8192
    temporal_filter_wmma<<<blocks, 256, 0, stream>>>(xs, xc, ft, tao, out);
}